// SparseSubdivideBlock3d_3813930959056
// MI455X (gfx1250) — compile-verified
//
#include <hip/hip_runtime.h>
#include <hip/hip_bf16.h>

typedef float v2f __attribute__((ext_vector_type(2)));
typedef float v8f __attribute__((ext_vector_type(8)));

#define RG 128            // subdivided resolution
#define GRID_CELLS (RG*RG*RG)

// ---------------- grid build ----------------
__global__ void k_grid_init(int* __restrict__ g, int n) {
    int i = blockIdx.x * 256 + threadIdx.x;
    if (i < n) g[i] = -1;
}

__global__ void k_grid_fill(const int* __restrict__ coords, int* __restrict__ g, int nchild) {
    int m = blockIdx.x * 256 + threadIdx.x;
    if (m >= nchild) return;
    int parent = m >> 3, child = m & 7;
    const int* cc = coords + parent * 4;
    int cx = (cc[1] << 1) + ((child >> 2) & 1);
    int cy = (cc[2] << 1) + ((child >> 1) & 1);
    int cz = (cc[3] << 1) + (child & 1);
    g[(cx * RG + cy) * RG + cz] = m;
}

// ---------------- group-norm stats (deterministic 2-stage) ----------------
__global__ __launch_bounds__(256)
void k_gn_stats(const float* __restrict__ x, int rows, float* __restrict__ partials) {
    __shared__ float s1[4][64];
    __shared__ float s2[4][64];
    int t = threadIdx.x;
    int c = t & 63, ln = t >> 6;
    float a = 0.f, b = 0.f;
    for (int r = blockIdx.x * 4 + ln; r < rows; r += gridDim.x * 4) {
        float v = x[(size_t)r * 64 + c];
        a += v; b += v * v;
    }
    s1[ln][c] = a; s2[ln][c] = b;
    __syncthreads();
    if (ln == 0) {
        a = s1[0][c] + s1[1][c] + s1[2][c] + s1[3][c];
        b = s2[0][c] + s2[1][c] + s2[2][c] + s2[3][c];
        partials[blockIdx.x * 128 + c]      = a;
        partials[blockIdx.x * 128 + 64 + c] = b;
    }
}

__global__ void k_gn_finalize(const float* __restrict__ partials, int nb, float rows,
                              const float* __restrict__ w, const float* __restrict__ bgn,
                              float* __restrict__ ss) {
    __shared__ float S[64];
    __shared__ float S2[64];
    int c = threadIdx.x;             // 64 threads
    float a = 0.f, b = 0.f;
    for (int i = 0; i < nb; ++i) {   // fixed order -> deterministic
        a += partials[i * 128 + c];
        b += partials[i * 128 + 64 + c];
    }
    S[c] = a; S2[c] = b;
    __syncthreads();
    int g = c >> 1;
    float cnt   = rows * 2.0f;       // C/G == 2 channels per group
    float sum   = S[2 * g] + S[2 * g + 1];
    float sumsq = S2[2 * g] + S2[2 * g + 1];
    float mu    = sum / cnt;
    float var   = sumsq / cnt - mu * mu;
    float rstd  = rsqrtf(var + 1e-5f);
    float sc    = w[c] * rstd;
    ss[c]       = sc;
    ss[64 + c]  = bgn[c] - mu * sc;
}

// ---------------- scale/shift + SiLU ----------------
__global__ void k_apply(const float* __restrict__ x, const float* __restrict__ ss,
                        float* __restrict__ y, int n) {
    int i = blockIdx.x * blockDim.x + threadIdx.x;
    if (i >= n) return;
    int c = i & 63;
    float v = x[i] * ss[c] + ss[64 + c];
    y[i] = v / (1.0f + expf(-v));
}

// ---------------- sparse 3x3x3 conv via f32 WMMA ----------------
// block = 128 threads (4 waves), owns 16 output rows x 64 cols.
// SHIFT=3: gather rows from parent table (8x repeat folded away). SKIP: add feats[m>>3].
template <int SHIFT, bool SKIP>
__global__ __launch_bounds__(128)
void k_conv(const float* __restrict__ src, const float* __restrict__ W,
            const float* __restrict__ bias, const float* __restrict__ skip,
            const int* __restrict__ coords, const int* __restrict__ grid,
            float* __restrict__ out) {
    __shared__ int   nbIdx[432];        // 27 offsets x 16 rows
    __shared__ float Atile[16][68];     // padded: conflict-free column reads

    const int m0 = blockIdx.x * 16;
    const int t  = threadIdx.x;

    // neighbor indices for all 27 offsets of our 16 rows
    for (int e = t; e < 432; e += 128) {
        int o = e >> 4, r = e & 15;
        int m = m0 + r;
        int parent = m >> 3, child = m & 7;
        const int* cc = coords + parent * 4;
        int cx = (cc[1] << 1) + ((child >> 2) & 1) + (o / 9) - 1;
        int cy = (cc[2] << 1) + ((child >> 1) & 1) + ((o / 3) % 3) - 1;
        int cz = (cc[3] << 1) + (child & 1) + (o % 3) - 1;
        int nb = -1;
        if ((unsigned)cx < (unsigned)RG && (unsigned)cy < (unsigned)RG &&
            (unsigned)cz < (unsigned)RG)
            nb = grid[(cx * RG + cy) * RG + cz];
        nbIdx[e] = nb;
    }
    __syncthreads();

    const int lane = t & 31;
    const int wave = t >> 5;
    const int d0   = wave * 16;        // this wave's 16-col tile
    const int col  = lane & 15;        // also A-fragment row (M = lane%16)
    const int hi   = lane >> 4;        // lanes 16-31 hold K+2 / rows M+8

    v8f acc = {};

    const int sr = t >> 3;             // staging row
    const int sc = (t & 7) * 8;        // staging col base (8 floats/thread)

    for (int o = 0; o < 27; ++o) {
        // stage gathered A tile (zeros for missing neighbors)
        {
            int nb = nbIdx[o * 16 + sr];
            float4 x0 = {0.f, 0.f, 0.f, 0.f}, x1 = {0.f, 0.f, 0.f, 0.f};
            if (nb >= 0) {
                const float* p = src + ((size_t)(nb >> SHIFT)) * 64 + sc;
                x0 = *(const float4*)p;
                x1 = *(const float4*)(p + 4);
            }
            *(float4*)&Atile[sr][sc]     = x0;
            *(float4*)&Atile[sr][sc + 4] = x1;
        }
        __syncthreads();

        const float* Wo = W + o * 4096 + d0 + col;   // W[o][c][d0+col]
        #pragma unroll
        for (int c0 = 0; c0 < 64; c0 += 4) {
            int ka = c0 + hi * 2;
            v2f a, b;
            a.x = Atile[col][ka];
            a.y = Atile[col][ka + 1];
            b.x = Wo[ka * 64];
            b.y = Wo[(ka + 1) * 64];
            acc = __builtin_amdgcn_wmma_f32_16x16x4_f32(
                false, a, false, b, (short)0, acc, false, false);
        }
        __syncthreads();
    }

    float bv = bias[d0 + col];
    #pragma unroll
    for (int v = 0; v < 8; ++v) {
        int row = v + hi * 8;
        int m = m0 + row;
        float val = acc[v] + bv;
        if (SKIP) val += skip[((size_t)(m >> 3)) * 64 + d0 + col];
        out[(size_t)m * 64 + d0 + col] = val;
    }
}

// ---------------- host launcher ----------------
extern "C" void kernel_launch(void* const* d_in, const int* in_sizes, int n_in,
                              void* d_out, int out_size, void* d_ws, size_t ws_size,
                              hipStream_t stream) {
    const int*   coords  = (const int*)d_in[0];
    const float* feats   = (const float*)d_in[1];
    const float* gn1_w   = (const float*)d_in[2];
    const float* gn1_b   = (const float*)d_in[3];
    const float* conv1_w = (const float*)d_in[4];
    const float* conv1_b = (const float*)d_in[5];
    const float* gn2_w   = (const float*)d_in[6];
    const float* gn2_b   = (const float*)d_in[7];
    const float* conv2_w = (const float*)d_in[8];
    const float* conv2_b = (const float*)d_in[9];
    float* out = (float*)d_out;

    const int N = in_sizes[0] / 4;   // active input voxels
    const int M = N * 8;             // children

    // workspace layout
    char*  ws       = (char*)d_ws;
    int*   grid     = (int*)ws;                                        // 8 MB
    float* partials = (float*)(ws + (size_t)GRID_CELLS * 4);           // 64 KB
    float* ss       = partials + 128 * 128;                            // 512 B
    float* h1       = ss + 128;                                        // N*64 f32
    float* h2       = h1 + (size_t)N * 64;                             // M*64 f32

    const int P1 = 64, P2 = 128;

    // build index grid
    k_grid_init<<<(GRID_CELLS + 255) / 256, 256, 0, stream>>>(grid, GRID_CELLS);
    k_grid_fill<<<(M + 255) / 256, 256, 0, stream>>>(coords, grid, M);

    // GN1 + SiLU -> h1
    k_gn_stats<<<P1, 256, 0, stream>>>(feats, N, partials);
    k_gn_finalize<<<1, 64, 0, stream>>>(partials, P1, (float)N, gn1_w, gn1_b, ss);
    k_apply<<<(N * 64 + 255) / 256, 256, 0, stream>>>(feats, ss, h1, N * 64);

    // conv1 (gather parent rows: nb>>3) -> d_out used as scratch
    k_conv<3, false><<<M / 16, 128, 0, stream>>>(h1, conv1_w, conv1_b, nullptr,
                                                 coords, grid, out);

    // GN2 + SiLU -> h2
    k_gn_stats<<<P2, 256, 0, stream>>>(out, M, partials);
    k_gn_finalize<<<1, 64, 0, stream>>>(partials, P2, (float)M, gn2_w, gn2_b, ss);
    k_apply<<<(M * 64 + 255) / 256, 256, 0, stream>>>(out, ss, h2, M * 64);

    // conv2 + identity skip -> d_out
    k_conv<0, true><<<M / 16, 128, 0, stream>>>(h2, conv2_w, conv2_b, feats,
                                                coords, grid, out);
}